// DomainBatchNormImpl_73443940761618
// MI455X (gfx1250) — compile-verified
//
#include <hip/hip_runtime.h>

#define N_SAMPLES 131072
#define C_CH      512
#define N_DOM     8
#define EPSV      1e-5f

typedef __attribute__((ext_vector_type(2))) float v2f;
typedef __attribute__((ext_vector_type(8))) float v8f;

// Workspace layout (float offsets)
#define WS_SUMS   0        // [8 x 512]
#define WS_SUMSQ  4096     // [8 x 512]
#define WS_CNT    8192     // [16] (8 used)
#define WS_SCALE  8208     // [8 x 512]
#define WS_SHIFT  12304    // [8 x 512]
// total 16400 floats = 65.6 KB

#define WMMA_F32X4 __builtin_amdgcn_wmma_f32_16x16x4_f32

// ---------------------------------------------------------------------------
// Pass 1: per-domain segment sums + sum-of-squares via one-hot WMMA GEMM.
//   D[m][n] += onehot(d[k]==m) * X[k][n]  with V_WMMA_F32_16X16X4_F32.
// Each wave owns TWO 16-channel groups (32 channels) and a 512-sample span:
// the A matrix (one-hot of d) is built once per 4-sample chunk and feeds
// 4 accumulator chains (sum/sumsq x 2 channel groups). Manual unroll x2
// chunks so 12 loads are issued ahead of 8 WMMAs per iteration, and all X
// addressing is one striding pointer + immediate offsets.
// ---------------------------------------------------------------------------
__global__ __launch_bounds__(256) void dbn_stats(const float* __restrict__ X,
                                                 const int*   __restrict__ dom,
                                                 float*       __restrict__ ws) {
  const int lane = threadIdx.x & 31;
  const int wave = threadIdx.x >> 5;
  const int cbp  = blockIdx.x;                 // 0..15 channel-pair (32 ch each)
  const int sg   = blockIdx.y * 8 + wave;      // 0..255 sample group
  const int c0   = cbp * 32;

  const int half = lane >> 4;                  // K split: lanes 0-15 -> K=0,1 ; 16-31 -> K=2,3
  const int lm   = lane & 15;                  // A: row M ; B: col N

  const int span = N_SAMPLES / 256;            // 512 samples per wave
  const int sBeg = sg * span;

  // Per-lane pointer to X[rA][c0 + lm]; strided by 8 rows per iteration.
  const float* p = X + (size_t)(sBeg + 2 * half) * C_CH + (size_t)(c0 + lm);
  const int*   q = dom + sBeg + 2 * half;

  v8f accS0 = {0.f,0.f,0.f,0.f,0.f,0.f,0.f,0.f};
  v8f accQ0 = {0.f,0.f,0.f,0.f,0.f,0.f,0.f,0.f};
  v8f accS1 = {0.f,0.f,0.f,0.f,0.f,0.f,0.f,0.f};
  v8f accQ1 = {0.f,0.f,0.f,0.f,0.f,0.f,0.f,0.f};
  int cnt = 0;

  for (int it = 0; it < span / 8; ++it) {
    // ---- hoisted loads: 2 chunks x (2 rows x 2 channel groups) + d pairs ----
    const int dA0 = q[0];
    const int dB0 = q[1];
    const int dA1 = q[4];
    const int dB1 = q[5];

    const float x00 = p[0];                    // chunk0 rA, group0
    const float x01 = p[C_CH];                 // chunk0 rB, group0
    const float x02 = p[16];                   // chunk0 rA, group1
    const float x03 = p[C_CH + 16];            // chunk0 rB, group1
    const float x10 = p[4 * C_CH];             // chunk1 rA, group0
    const float x11 = p[5 * C_CH];             // chunk1 rB, group0
    const float x12 = p[4 * C_CH + 16];        // chunk1 rA, group1
    const float x13 = p[5 * C_CH + 16];        // chunk1 rB, group1

    // ---- one-hot A matrices (rows 8..15 stay zero: domains are 0..7) ----
    v2f a0, a1;
    a0.x = (dA0 == lm) ? 1.0f : 0.0f;
    a0.y = (dB0 == lm) ? 1.0f : 0.0f;
    a1.x = (dA1 == lm) ? 1.0f : 0.0f;
    a1.y = (dB1 == lm) ? 1.0f : 0.0f;

    v2f b00; b00.x = x00;       b00.y = x01;
    v2f q00; q00.x = x00 * x00; q00.y = x01 * x01;
    v2f b01; b01.x = x02;       b01.y = x03;
    v2f q01; q01.x = x02 * x02; q01.y = x03 * x03;
    v2f b10; b10.x = x10;       b10.y = x11;
    v2f q10; q10.x = x10 * x10; q10.y = x11 * x11;
    v2f b11; b11.x = x12;       b11.y = x13;
    v2f q11; q11.x = x12 * x12; q11.y = x13 * x13;

    // ---- 8 WMMAs per iteration ----
    accS0 = WMMA_F32X4(false, a0, false, b00, (short)0, accS0, false, false);
    accQ0 = WMMA_F32X4(false, a0, false, q00, (short)0, accQ0, false, false);
    accS1 = WMMA_F32X4(false, a0, false, b01, (short)0, accS1, false, false);
    accQ1 = WMMA_F32X4(false, a0, false, q01, (short)0, accQ1, false, false);
    accS0 = WMMA_F32X4(false, a1, false, b10, (short)0, accS0, false, false);
    accQ0 = WMMA_F32X4(false, a1, false, q10, (short)0, accQ0, false, false);
    accS1 = WMMA_F32X4(false, a1, false, b11, (short)0, accS1, false, false);
    accQ1 = WMMA_F32X4(false, a1, false, q11, (short)0, accQ1, false, false);

    cnt += (dA0 == lm) + (dB0 == lm) + (dA1 == lm) + (dB1 == lm);

    p += 8 * C_CH;
    q += 8;
  }

  // Domain counts: only one channel-pair needs to histogram d.
  if (cbp == 0 && lm < N_DOM) {
    atomicAdd(&ws[WS_CNT + lm], (float)cnt);
  }

  // C/D layout: VGPR v, lanes 0-15 -> row M=v (real domains 0..7), col N=lane.
  // Lanes 16-31 hold padded rows 8..15 (identically zero) -> skip.
  if (half == 0) {
    float* sums  = ws + WS_SUMS;
    float* sumsq = ws + WS_SUMSQ;
    const int cA = c0 + lm;        // group0 column
    const int cB = c0 + 16 + lm;   // group1 column
#pragma unroll
    for (int v = 0; v < 8; ++v) {
      atomicAdd(&sums [v * C_CH + cA], accS0[v]);
      atomicAdd(&sums [v * C_CH + cB], accS1[v]);
      atomicAdd(&sumsq[v * C_CH + cA], accQ0[v]);
      atomicAdd(&sumsq[v * C_CH + cB], accQ1[v]);
    }
  }
}

// ---------------------------------------------------------------------------
// Pass 1b: fold stats + shared affine params into per-(domain,channel)
// scale/shift so the streaming pass is a single FMA per element.
//   mean = sum/cnt ; var = (sumsq - sum*mean)/max(cnt-1,1) (unbiased)
//   out = X*scale + shift,  scale = rsqrt(var+eps)*std_p,  shift = mean_p - mean*scale
// ---------------------------------------------------------------------------
__global__ __launch_bounds__(256) void dbn_finalize(const float* __restrict__ mean_p,
                                                    const float* __restrict__ std_p,
                                                    float*       __restrict__ ws) {
  const int i = blockIdx.x * blockDim.x + threadIdx.x;   // 0..4095
  if (i >= N_DOM * C_CH) return;
  const int m = i / C_CH;
  const int c = i - m * C_CH;

  const float cntf = ws[WS_CNT + m];
  const float cnt  = fmaxf(cntf, 1.0f);
  const float sum  = ws[WS_SUMS + i];
  const float sq   = ws[WS_SUMSQ + i];

  const float mean = sum / cnt;
  float var = (sq - sum * mean) / fmaxf(cntf - 1.0f, 1.0f);
  var = fmaxf(var, 0.0f);
  const float inv = rsqrtf(var + EPSV);

  const float s = inv * std_p[c];
  ws[WS_SCALE + i] = s;
  ws[WS_SHIFT + i] = mean_p[c] - mean * s;
}

// ---------------------------------------------------------------------------
// Pass 2: streaming normalize, float4 vectorized. 512 MB of traffic at full
// HBM rate; 32 KB scale/shift tables stay resident in cache.
// ---------------------------------------------------------------------------
__global__ __launch_bounds__(256) void dbn_apply(const float* __restrict__ X,
                                                 const int*   __restrict__ dom,
                                                 const float* __restrict__ ws,
                                                 float*       __restrict__ out) {
  const size_t gid = (size_t)blockIdx.x * blockDim.x + threadIdx.x; // one float4 each
  const int row = (int)(gid >> 7);          // C/4 = 128 float4 per row
  const int c4  = (int)(gid & 127);
  const int m   = dom[row];

  const float4* x4 = (const float4*)X;
  const float4* s4 = (const float4*)(ws + WS_SCALE);
  const float4* b4 = (const float4*)(ws + WS_SHIFT);

  const float4 x = x4[gid];
  const float4 s = s4[m * 128 + c4];
  const float4 b = b4[m * 128 + c4];

  float4 o;
  o.x = fmaf(x.x, s.x, b.x);
  o.y = fmaf(x.y, s.y, b.y);
  o.z = fmaf(x.z, s.z, b.z);
  o.w = fmaf(x.w, s.w, b.w);
  ((float4*)out)[gid] = o;
}

extern "C" void kernel_launch(void* const* d_in, const int* in_sizes, int n_in,
                              void* d_out, int out_size, void* d_ws, size_t ws_size,
                              hipStream_t stream) {
  const float* X      = (const float*)d_in[0];
  const int*   dom    = (const int*)  d_in[1];
  const float* mean_p = (const float*)d_in[2];
  const float* std_p  = (const float*)d_in[3];
  float*       ws     = (float*)d_ws;
  float*       out    = (float*)d_out;

  // Zero the accumulator region (sums + sumsq + counts) every call.
  hipMemsetAsync(ws, 0, (size_t)(WS_CNT + 16) * sizeof(float), stream);

  // Pass 1: stats. 16 channel-pairs x 32 blocks x 8 waves = 4096 waves,
  // each wave: 512 samples, 64 iterations x 8 V_WMMA_F32_16X16X4_F32.
  dim3 gs(16, 32);
  dbn_stats<<<gs, 256, 0, stream>>>(X, dom, ws);

  // Finalize 8x512 scale/shift.
  dbn_finalize<<<16, 256, 0, stream>>>(mean_p, std_p, ws);

  // Pass 2: streaming normalize. 131072*512/4 float4 elements.
  dbn_apply<<<(N_SAMPLES * (C_CH / 4)) / 256, 256, 0, stream>>>(X, dom, ws, out);
}